// GraphTransformer_64957085385319
// MI455X (gfx1250) — compile-verified
//
#include <hip/hip_runtime.h>

typedef __attribute__((ext_vector_type(2))) float v2f;
typedef __attribute__((ext_vector_type(8))) float v8f;

#define HID 128
#define LN_EPS 1e-5f

// ---------------------------------------------------------------------------
// C[N,128] = A[N,128] @ W[128,128] + bias[128]
// Block: 256 threads (8 waves), computes 128 rows. W cached in LDS (64KB).
// Biases preloaded to registers (no VMEM waits inside the tile loop).
// ---------------------------------------------------------------------------
__global__ __launch_bounds__(256) void gemm128_bias(
    const float* __restrict__ A, const float* __restrict__ W,
    const float* __restrict__ bias, float* __restrict__ C, int N)
{
    __shared__ float Wlds[HID * HID];   // 64 KB
    {
        const float4* Wv = (const float4*)W;
        float4*       Lv = (float4*)Wlds;
        #pragma unroll
        for (int i = 0; i < (HID * HID / 4) / 256; ++i)
            Lv[i * 256 + threadIdx.x] = Wv[i * 256 + threadIdx.x];
    }

    const int lane = threadIdx.x & 31;
    const int wid  = threadIdx.x >> 5;
    const int half = lane >> 4;        // lanes 16-31 hold K+2 / M+8
    const int l16  = lane & 15;
    const int row0 = blockIdx.x * 128 + wid * 16;

    int arow = row0 + l16;
    if (arow > N - 1) arow = N - 1;    // clamp loads; stores guarded below

    // Preload biases + A fragments while LDS fill completes.
    float bvals[8];
    #pragma unroll
    for (int c = 0; c < 8; ++c) bvals[c] = bias[c * 16 + l16];

    v2f afrag[32];
    const float* Ap = A + (long)arow * HID + 2 * half;
    #pragma unroll
    for (int kk = 0; kk < 32; ++kk)
        afrag[kk] = *(const v2f*)(Ap + kk * 4);

    __syncthreads();

    for (int c = 0; c < 8; ++c) {
        const int n = c * 16 + l16;    // output column for this lane
        const float bv = bvals[c];
        v8f acc = {bv, bv, bv, bv, bv, bv, bv, bv};
        #pragma unroll
        for (int kk = 0; kk < 32; ++kk) {
            const int krow = kk * 4 + 2 * half;
            v2f bfrag;
            bfrag.x = Wlds[krow * HID + n];
            bfrag.y = Wlds[(krow + 1) * HID + n];
            acc = __builtin_amdgcn_wmma_f32_16x16x4_f32(
                false, afrag[kk], false, bfrag, (short)0, acc, false, false);
        }
        #pragma unroll
        for (int j = 0; j < 8; ++j) {
            const int m = row0 + j + 8 * half;
            if (m < N) C[(long)m * HID + n] = acc[j];
        }
    }
}

// ---------------------------------------------------------------------------
// Fused Q/K/V projection: three 128x128 weight matrices resident in 192 KB
// of dynamic LDS (gfx1250 WGP has 320 KB). A-fragments loaded once, reused
// for all three WMMA streams.
// ---------------------------------------------------------------------------
__global__ __launch_bounds__(256) void gemm128_qkv(
    const float* __restrict__ A,
    const float* __restrict__ WQ, const float* __restrict__ bQ,
    const float* __restrict__ WK, const float* __restrict__ bK,
    const float* __restrict__ WV, const float* __restrict__ bV,
    float* __restrict__ Qo, float* __restrict__ Ko, float* __restrict__ Vo,
    int N)
{
    extern __shared__ float smem[];            // 3 * 64 KB = 192 KB
    float* Wq = smem;
    float* Wk = smem + HID * HID;
    float* Wv = smem + 2 * HID * HID;
    {
        const int nv4 = (HID * HID) / 4;       // 4096 float4 per matrix
        const float4* q4 = (const float4*)WQ;
        const float4* k4 = (const float4*)WK;
        const float4* v4 = (const float4*)WV;
        #pragma unroll
        for (int i = 0; i < nv4 / 256; ++i) {
            const int idx = i * 256 + threadIdx.x;
            ((float4*)Wq)[idx] = q4[idx];
            ((float4*)Wk)[idx] = k4[idx];
            ((float4*)Wv)[idx] = v4[idx];
        }
    }

    const int lane = threadIdx.x & 31;
    const int wid  = threadIdx.x >> 5;
    const int half = lane >> 4;
    const int l16  = lane & 15;
    const int row0 = blockIdx.x * 128 + wid * 16;

    int arow = row0 + l16;
    if (arow > N - 1) arow = N - 1;

    float bq[8], bk[8], bv[8];
    #pragma unroll
    for (int c = 0; c < 8; ++c) {
        bq[c] = bQ[c * 16 + l16];
        bk[c] = bK[c * 16 + l16];
        bv[c] = bV[c * 16 + l16];
    }

    v2f afrag[32];
    const float* Ap = A + (long)arow * HID + 2 * half;
    #pragma unroll
    for (int kk = 0; kk < 32; ++kk)
        afrag[kk] = *(const v2f*)(Ap + kk * 4);

    __syncthreads();

    for (int c = 0; c < 8; ++c) {
        const int n = c * 16 + l16;
        v8f accq = {bq[c], bq[c], bq[c], bq[c], bq[c], bq[c], bq[c], bq[c]};
        v8f acck = {bk[c], bk[c], bk[c], bk[c], bk[c], bk[c], bk[c], bk[c]};
        v8f accv = {bv[c], bv[c], bv[c], bv[c], bv[c], bv[c], bv[c], bv[c]};
        #pragma unroll
        for (int kk = 0; kk < 32; ++kk) {
            const int krow = kk * 4 + 2 * half;
            v2f fq, fk, fv;
            fq.x = Wq[krow * HID + n];  fq.y = Wq[(krow + 1) * HID + n];
            fk.x = Wk[krow * HID + n];  fk.y = Wk[(krow + 1) * HID + n];
            fv.x = Wv[krow * HID + n];  fv.y = Wv[(krow + 1) * HID + n];
            accq = __builtin_amdgcn_wmma_f32_16x16x4_f32(
                false, afrag[kk], false, fq, (short)0, accq, false, false);
            acck = __builtin_amdgcn_wmma_f32_16x16x4_f32(
                false, afrag[kk], false, fk, (short)0, acck, false, false);
            accv = __builtin_amdgcn_wmma_f32_16x16x4_f32(
                false, afrag[kk], false, fv, (short)0, accv, false, false);
        }
        #pragma unroll
        for (int j = 0; j < 8; ++j) {
            const int m = row0 + j + 8 * half;
            if (m < N) {
                Qo[(long)m * HID + n] = accq[j];
                Ko[(long)m * HID + n] = acck[j];
                Vo[(long)m * HID + n] = accv[j];
            }
        }
    }
}

// ---------------------------------------------------------------------------
// Edge scores + segment max.  One wave per edge: 32 lanes x float4 dot.
// Float max via ordered-uint atomicMax (memset 0 == -inf key).
// ---------------------------------------------------------------------------
__device__ __forceinline__ unsigned f2key(float f) {
    unsigned u = __float_as_uint(f);
    return (u & 0x80000000u) ? ~u : (u | 0x80000000u);
}
__device__ __forceinline__ float key2f(unsigned k) {
    unsigned u = (k & 0x80000000u) ? (k & 0x7fffffffu) : ~k;
    return __uint_as_float(u);
}

__global__ __launch_bounds__(256) void edge_scores_max(
    const float* __restrict__ Q, const float* __restrict__ K,
    const long long* __restrict__ src, const long long* __restrict__ dst,
    float* __restrict__ scores, unsigned* __restrict__ segmax, int E)
{
    const int lane = threadIdx.x & 31;
    const int e = blockIdx.x * 8 + (threadIdx.x >> 5);
    if (e >= E) return;
    const int s = (int)src[e];
    const int d = (int)dst[e];
    const float4 q = *(const float4*)(Q + (long)d * HID + lane * 4);
    const float4 k = *(const float4*)(K + (long)s * HID + lane * 4);
    float p = q.x * k.x + q.y * k.y + q.z * k.z + q.w * k.w;
    #pragma unroll
    for (int off = 16; off > 0; off >>= 1) p += __shfl_xor(p, off, 32);
    if (lane == 0) {
        const float sc = p * 0.08838834764831845f;  // 1/sqrt(128)
        scores[e] = sc;
        atomicMax(segmax + d, f2key(sc));
    }
}

// ---------------------------------------------------------------------------
// e = exp(score - segmax[dst]); segment sum.  One thread per edge.
// ---------------------------------------------------------------------------
__global__ __launch_bounds__(256) void edge_exp_sum(
    const long long* __restrict__ dst, float* __restrict__ scores,
    const unsigned* __restrict__ segmax, float* __restrict__ segsum, int E)
{
    const int e = blockIdx.x * 256 + threadIdx.x;
    if (e >= E) return;
    const int d = (int)dst[e];
    const float ev = __expf(scores[e] - key2f(segmax[d]));
    scores[e] = ev;   // overwrite with exp value
    atomicAdd(segsum + d, ev);
}

// ---------------------------------------------------------------------------
// msg[dst] += (e/segsum[dst]) * V[src].  One wave per edge, 4 feats/lane.
// ---------------------------------------------------------------------------
__global__ __launch_bounds__(256) void edge_messages(
    const float* __restrict__ V, const long long* __restrict__ src,
    const long long* __restrict__ dst, const float* __restrict__ escore,
    const float* __restrict__ segsum, float* __restrict__ msg, int E)
{
    const int lane = threadIdx.x & 31;
    const int e = blockIdx.x * 8 + (threadIdx.x >> 5);
    if (e >= E) return;
    const int s = (int)src[e];
    const int d = (int)dst[e];
    const float attn = escore[e] / segsum[d];
    const float4 v = *(const float4*)(V + (long)s * HID + lane * 4);
    float* out = msg + (long)d * HID + lane * 4;
    atomicAdd(out + 0, attn * v.x);
    atomicAdd(out + 1, attn * v.y);
    atomicAdd(out + 2, attn * v.z);
    atomicAdd(out + 3, attn * v.w);
}

// ---------------------------------------------------------------------------
// h = LayerNorm(h + o) * g + b.  One wave per node, 4 elems/lane.
// ---------------------------------------------------------------------------
__global__ __launch_bounds__(256) void residual_ln(
    const float* __restrict__ h, const float* __restrict__ o,
    const float* __restrict__ g, const float* __restrict__ bta,
    float* __restrict__ hout, int N)
{
    const int lane = threadIdx.x & 31;
    const int node = blockIdx.x * 8 + (threadIdx.x >> 5);
    if (node >= N) return;
    const float4 a = *(const float4*)(h + (long)node * HID + lane * 4);
    const float4 c = *(const float4*)(o + (long)node * HID + lane * 4);
    float4 y = {a.x + c.x, a.y + c.y, a.z + c.z, a.w + c.w};
    float s = y.x + y.y + y.z + y.w;
    #pragma unroll
    for (int off = 16; off > 0; off >>= 1) s += __shfl_xor(s, off, 32);
    const float mu = s * (1.0f / 128.0f);
    const float dx = y.x - mu, dy = y.y - mu, dz = y.z - mu, dw = y.w - mu;
    float v = dx * dx + dy * dy + dz * dz + dw * dw;
    #pragma unroll
    for (int off = 16; off > 0; off >>= 1) v += __shfl_xor(v, off, 32);
    const float inv = rsqrtf(v * (1.0f / 128.0f) + LN_EPS);
    const float4 gg = *(const float4*)(g + lane * 4);
    const float4 bb = *(const float4*)(bta + lane * 4);
    float4 r = {dx * inv * gg.x + bb.x, dy * inv * gg.y + bb.y,
                dz * inv * gg.z + bb.z, dw * inv * gg.w + bb.w};
    *(float4*)(hout + (long)node * HID + lane * 4) = r;
}

// ---------------------------------------------------------------------------
extern "C" void kernel_launch(void* const* d_in, const int* in_sizes, int n_in,
                              void* d_out, int out_size, void* d_ws, size_t ws_size,
                              hipStream_t stream)
{
    (void)n_in; (void)out_size; (void)ws_size;
    const float*     x    = (const float*)d_in[0];
    const long long* edge = (const long long*)d_in[1];   // int64 [2,E]
    const float*     Wn   = (const float*)d_in[2];
    const float*     bn   = (const float*)d_in[3];
    const float*     WQ   = (const float*)d_in[4];
    const float*     bQ   = (const float*)d_in[5];
    const float*     WK   = (const float*)d_in[6];
    const float*     bK   = (const float*)d_in[7];
    const float*     WV   = (const float*)d_in[8];
    const float*     bV   = (const float*)d_in[9];
    const float*     WO   = (const float*)d_in[10];
    const float*     bO   = (const float*)d_in[11];
    const float*     lng  = (const float*)d_in[12];
    const float*     lnb  = (const float*)d_in[13];
    const float*     Wout = (const float*)d_in[14];
    const float*     bout = (const float*)d_in[15];

    const int N = in_sizes[0] / HID;
    const int E = in_sizes[1] / 2;
    const long long* src = edge;
    const long long* dst = edge + E;

    char* ws = (char*)d_ws;
    const size_t nh = (size_t)N * HID * sizeof(float);
    float*    h    = (float*)(ws);
    float*    Qb   = (float*)(ws + nh);          // reused as O-proj output
    float*    Kb   = (float*)(ws + 2 * nh);
    float*    Vb   = (float*)(ws + 3 * nh);
    float*    msg  = (float*)(ws + 4 * nh);
    float*    sco  = (float*)(ws + 5 * nh);
    unsigned* smax = (unsigned*)(ws + 5 * nh + (size_t)E * 4);
    float*    ssum = (float*)(ws + 5 * nh + (size_t)E * 4 + (size_t)N * 4);

    const int gemmBlk  = (N + 127) / 128;
    const int edgeWBlk = (E + 7) / 8;
    const int edgeTBlk = (E + 255) / 256;
    const int nodeWBlk = (N + 7) / 8;
    const size_t qkvLds = 3u * HID * HID * sizeof(float);   // 192 KB

    gemm128_bias<<<gemmBlk, 256, 0, stream>>>(x, Wn, bn, h, N);

    for (int l = 0; l < 4; ++l) {
        const size_t wo = (size_t)l * HID * HID;
        const size_t bo = (size_t)l * HID;

        gemm128_qkv<<<gemmBlk, 256, qkvLds, stream>>>(
            h, WQ + wo, bQ + bo, WK + wo, bK + bo, WV + wo, bV + bo,
            Qb, Kb, Vb, N);

        hipMemsetAsync(smax, 0, (size_t)N * 4, stream);
        hipMemsetAsync(ssum, 0, (size_t)N * 4, stream);
        hipMemsetAsync(msg,  0, nh,            stream);

        edge_scores_max<<<edgeWBlk, 256, 0, stream>>>(Qb, Kb, src, dst, sco, smax, E);
        edge_exp_sum  <<<edgeTBlk, 256, 0, stream>>>(dst, sco, smax, ssum, E);
        edge_messages <<<edgeWBlk, 256, 0, stream>>>(Vb, src, dst, sco, ssum, msg, E);

        gemm128_bias<<<gemmBlk, 256, 0, stream>>>(msg, WO + wo, bO + bo, Qb, N);
        residual_ln <<<nodeWBlk, 256, 0, stream>>>(h, Qb, lng + bo, lnb + bo, h, N);
    }

    gemm128_bias<<<gemmBlk, 256, 0, stream>>>(h, Wout, bout, (float*)d_out, N);
}